// GraphEEGClassifier_22711787061812
// MI455X (gfx1250) — compile-verified
//
#include <hip/hip_runtime.h>

typedef float v2f __attribute__((ext_vector_type(2)));
typedef float v8f __attribute__((ext_vector_type(8)));

#define F3 64
#define NCLS 4
#define BN_EPS 1e-5f

// ---------------- utility ----------------
__global__ void zero_f32(float* __restrict__ p, long long n) {
  long long i = (long long)blockIdx.x * blockDim.x + threadIdx.x;
  long long stride = (long long)gridDim.x * blockDim.x;
  for (; i < n; i += stride) p[i] = 0.0f;
}

// ---------------- gcn_norm ----------------
__global__ void deg_kernel(const int* __restrict__ col, const float* __restrict__ w,
                           float* __restrict__ deg, long long E) {
  long long e = (long long)blockIdx.x * blockDim.x + threadIdx.x;
  if (e < E) atomicAdd(&deg[col[e]], w[e]);
}

__global__ void dis_kernel(float* __restrict__ deg, long long N) {
  long long i = (long long)blockIdx.x * blockDim.x + threadIdx.x;
  if (i < N) {
    float d = deg[i];
    deg[i] = (d > 0.0f) ? rsqrtf(fmaxf(d, 1e-30f)) : 0.0f;
  }
}

__global__ void norm_kernel(const int* __restrict__ row, const int* __restrict__ col,
                            const float* __restrict__ w, const float* __restrict__ dis,
                            float* __restrict__ nrm, long long E) {
  long long e = (long long)blockIdx.x * blockDim.x + threadIdx.x;
  if (e < E) nrm[e] = dis[row[e]] * w[e] * dis[col[e]];
}

// ---------------- dense GEMM via V_WMMA_F32_16X16X4_F32 ----------------
// H[n][f] = sum_k X[n][k] * W[f][k]   (i.e. H = X @ W^T), F3 = 64.
// One wave -> one 16x16 tile of H; block of 128 threads (4 waves) covers
// the 4 column tiles of a 16-row strip. K accumulated in 16 steps of 4.
__global__ void gemm_xwT_wmma(const float* __restrict__ X, const float* __restrict__ W,
                              float* __restrict__ H, int nrows) {
  const int lane    = threadIdx.x & 31;
  const int wave    = threadIdx.x >> 5;      // 0..3 -> column tile
  const int rowBase = blockIdx.x * 16;
  const int colBase = wave * 16;
  const int half    = lane >> 4;             // 0: K pair {0,1}, 1: K pair {2,3}
  const int l16     = lane & 15;

  int mrow = rowBase + l16;
  if (mrow >= nrows) mrow = nrows - 1;       // clamp loads; EXEC stays all-1s
  const int ncol = colBase + l16;
  const int kA   = half * 2;

  const float* __restrict__ xrow = X + (long long)mrow * F3;
  const float* __restrict__ wrow = W + (long long)ncol * F3;   // B[k][n] = W[n][k]

  v8f c = {0.f, 0.f, 0.f, 0.f, 0.f, 0.f, 0.f, 0.f};
#pragma unroll
  for (int k0 = 0; k0 < F3; k0 += 4) {
    v2f a, b;
    a.x = xrow[k0 + kA];
    a.y = xrow[k0 + kA + 1];
    b.x = wrow[k0 + kA];
    b.y = wrow[k0 + kA + 1];
    // (neg_a, A, neg_b, B, c_mod, C, reuse_a, reuse_b)
    c = __builtin_amdgcn_wmma_f32_16x16x4_f32(false, a, false, b, (short)0, c,
                                              false, false);
  }
  // C/D layout: VGPR r -> row (r + half*8), lane l16 -> column
#pragma unroll
  for (int r = 0; r < 8; ++r) {
    int orow = rowBase + r + half * 8;
    if (orow < nrows) H[(long long)orow * F3 + colBase + l16] = c[r];
  }
}

// ---------------- edge message scatter-add ----------------
// out[col[e]][f] += nrm[e] * H[row[e]][f]; 64 lanes per edge.
__global__ void scatter_kernel(const float* __restrict__ H, const int* __restrict__ row,
                               const int* __restrict__ col, const float* __restrict__ nrm,
                               float* __restrict__ out, long long E) {
  long long idx = (long long)blockIdx.x * blockDim.x + threadIdx.x;
  long long e = idx >> 6;
  int f = (int)(idx & 63);
  if (e >= E) return;
  int r = row[e], c = col[e];
  float v = nrm[e] * H[(long long)r * F3 + f];
  atomicAdd(&out[(long long)c * F3 + f], v);
}

// ---------------- eval BatchNorm + ELU ----------------
__global__ void bnelu_kernel(const float* __restrict__ agg,
                             const float* __restrict__ gamma, const float* __restrict__ beta,
                             const float* __restrict__ rm, const float* __restrict__ rv,
                             float* __restrict__ out, long long total) {
  long long i = (long long)blockIdx.x * blockDim.x + threadIdx.x;
  if (i >= total) return;
  int f = (int)(i & 63);
  float v = (agg[i] - rm[f]) * rsqrtf(rv[f] + BN_EPS) * gamma[f] + beta[f];
  out[i] = (v > 0.0f) ? v : expm1f(v);
}

// ---------------- global mean pool (sums + counts) ----------------
__global__ void pool_sum_kernel(const float* __restrict__ x, const int* __restrict__ batch,
                                float* __restrict__ pooled, long long N) {
  long long idx = (long long)blockIdx.x * blockDim.x + threadIdx.x;
  long long n = idx >> 6;
  int f = (int)(idx & 63);
  if (n >= N) return;
  atomicAdd(&pooled[(long long)batch[n] * F3 + f], x[n * F3 + f]);
}

__global__ void pool_cnt_kernel(const int* __restrict__ batch, float* __restrict__ cnt,
                                long long N) {
  long long n = (long long)blockIdx.x * blockDim.x + threadIdx.x;
  if (n < N) atomicAdd(&cnt[batch[n]], 1.0f);
}

// ---------------- classifier head: out = (pooled/cnt) @ lnW^T + lnb ----------------
__global__ void head_kernel(const float* __restrict__ pooled, const float* __restrict__ cnt,
                            const float* __restrict__ lnW, const float* __restrict__ lnb,
                            float* __restrict__ out, int G) {
  int idx = blockIdx.x * blockDim.x + threadIdx.x;
  if (idx >= G * NCLS) return;
  int g = idx / NCLS;
  int c = idx % NCLS;
  float inv = 1.0f / fmaxf(cnt[g], 1.0f);
  float acc = 0.0f;
#pragma unroll
  for (int k = 0; k < F3; ++k)
    acc += pooled[(long long)g * F3 + k] * inv * lnW[c * F3 + k];
  out[idx] = acc + lnb[c];
}

// ---------------- launcher ----------------
extern "C" void kernel_launch(void* const* d_in, const int* in_sizes, int n_in,
                              void* d_out, int out_size, void* d_ws, size_t ws_size,
                              hipStream_t stream) {
  const float* x     = (const float*)d_in[0];
  const int*   ei    = (const int*)d_in[1];
  const float* ew    = (const float*)d_in[2];
  const int*   batch = (const int*)d_in[3];
  const float* W1    = (const float*)d_in[4];
  const float* g1    = (const float*)d_in[5];
  const float* b1    = (const float*)d_in[6];
  const float* rm1   = (const float*)d_in[7];
  const float* rv1   = (const float*)d_in[8];
  const float* W2    = (const float*)d_in[9];
  const float* g2    = (const float*)d_in[10];
  const float* b2    = (const float*)d_in[11];
  const float* rm2   = (const float*)d_in[12];
  const float* rv2   = (const float*)d_in[13];
  const float* lnW   = (const float*)d_in[14];
  const float* lnb   = (const float*)d_in[15];

  const long long N = in_sizes[0] / F3;
  const long long E = in_sizes[2];
  const int G = out_size / NCLS;

  const int* erow = ei;        // edge_index[0]
  const int* ecol = ei + E;    // edge_index[1]

  // workspace carving (256B aligned slices)
  char* ws = (char*)d_ws;
  size_t off = 0;
  auto take = [&](size_t bytes) {
    char* p = ws + off;
    off = (off + bytes + 255) & ~(size_t)255;
    return p;
  };
  float* deg    = (float*)take((size_t)N * 4);         // then holds dis (in place)
  float* nrm    = (float*)take((size_t)E * 4);
  float* bufA   = (float*)take((size_t)N * F3 * 4);
  float* bufB   = (float*)take((size_t)N * F3 * 4);
  float* pooled = (float*)take((size_t)G * F3 * 4);
  float* cnt    = (float*)take((size_t)G * 4);
  (void)ws_size;

  const int T = 256;
  const long long NE  = N * F3;            // node-feature elements
  const long long EF  = E * F3;            // edge-feature work items
  const int zeroBlocks = 4096;             // grid-stride zero kernel

  // 1) gcn_norm
  zero_f32<<<zeroBlocks, T, 0, stream>>>(deg, N);
  deg_kernel<<<(int)((E + T - 1) / T), T, 0, stream>>>(ecol, ew, deg, E);
  dis_kernel<<<(int)((N + T - 1) / T), T, 0, stream>>>(deg, N);
  norm_kernel<<<(int)((E + T - 1) / T), T, 0, stream>>>(erow, ecol, ew, deg, nrm, E);

  const int gemmBlocks = (int)((N + 15) / 16);   // 16 rows per block, 4 waves/block

  // 2) layer 1: A = x @ W1^T (WMMA); B = scatter(A); A = BN+ELU(B)
  gemm_xwT_wmma<<<gemmBlocks, 128, 0, stream>>>(x, W1, bufA, (int)N);
  zero_f32<<<zeroBlocks, T, 0, stream>>>(bufB, NE);
  scatter_kernel<<<(int)((EF + T - 1) / T), T, 0, stream>>>(bufA, erow, ecol, nrm, bufB, E);
  bnelu_kernel<<<(int)((NE + T - 1) / T), T, 0, stream>>>(bufB, g1, b1, rm1, rv1, bufA, NE);

  // 3) layer 2: B = A @ W2^T (WMMA); A = scatter(B); B = BN+ELU(A)
  gemm_xwT_wmma<<<gemmBlocks, 128, 0, stream>>>(bufA, W2, bufB, (int)N);
  zero_f32<<<zeroBlocks, T, 0, stream>>>(bufA, NE);
  scatter_kernel<<<(int)((EF + T - 1) / T), T, 0, stream>>>(bufB, erow, ecol, nrm, bufA, E);
  bnelu_kernel<<<(int)((NE + T - 1) / T), T, 0, stream>>>(bufA, g2, b2, rm2, rv2, bufB, NE);

  // 4) mean pool + head
  zero_f32<<<zeroBlocks, T, 0, stream>>>(pooled, (long long)G * F3);
  zero_f32<<<zeroBlocks, T, 0, stream>>>(cnt, (long long)G);
  pool_sum_kernel<<<(int)((NE + T - 1) / T), T, 0, stream>>>(bufB, batch, pooled, N);
  pool_cnt_kernel<<<(int)((N + T - 1) / T), T, 0, stream>>>(batch, cnt, N);
  head_kernel<<<(G * NCLS + T - 1) / T, T, 0, stream>>>(pooled, cnt, lnW, lnb,
                                                        (float*)d_out, G);
}